// VariationalGraphConvEncoder_89635967467604
// MI455X (gfx1250) — compile-verified
//
#include <hip/hip_runtime.h>
#include <hip/hip_bf16.h>

// ---------------------------------------------------------------------------
// VGAE encoder (GCNConv x3) for MI455X / gfx1250 (wave32, WMMA).
// Memory-bound workload: GEMMs (4 GFLOP) run on V_WMMA_F32_16X16X4_F32 at
// full fp32 precision; edge aggregation (~1.3 GB traffic) uses per-wave
// gather + global_atomic_add_f32 scatter that resolves in the 192 MB L2.
// ---------------------------------------------------------------------------

typedef float v2f __attribute__((ext_vector_type(2)));
typedef float v8f __attribute__((ext_vector_type(8)));

#define CH_IN  100
#define CH_HID 100
#define CH_OUT 50

// ---------------- degree / normalization ----------------

__global__ void vgae_init_deg(float* deg, int n) {
    int i = blockIdx.x * blockDim.x + threadIdx.x;
    if (i < n) deg[i] = 1.0f;  // self-loop
}

__global__ void vgae_accum_deg(const int* __restrict__ dst, float* deg, int e) {
    int i = blockIdx.x * blockDim.x + threadIdx.x;
    if (i < e) atomicAdd(&deg[dst[i]], 1.0f);
}

__global__ void vgae_rsqrt(float* deg, int n) {
    int i = blockIdx.x * blockDim.x + threadIdx.x;
    if (i < n) deg[i] = rsqrtf(deg[i]);
}

// ---------------- fp32 WMMA GEMM: C[M x N] = A[M x K] @ B[K x N] ------------
// One wave per 16-row M-tile; NT 16-wide N-tiles held in accumulators.
// Requires M % 16 == 0 and K % 4 == 0 (holds here: M=100000, K=100).

template <int NT>
__global__ __launch_bounds__(128) void vgae_gemm_wmma_f32(
    const float* __restrict__ A, const float* __restrict__ B,
    float* __restrict__ C, int M, int N, int K) {
    const int wave  = threadIdx.x >> 5;
    const int lane  = threadIdx.x & 31;
    const int mtile = blockIdx.x * 4 + wave;
    if (mtile * 16 >= M) return;              // wave-uniform: EXEC stays full

    const int mbase = mtile * 16;
    const int nrow  = lane & 15;              // M index (A) / N index (B,C)
    const int koff  = (lane >> 4) << 1;       // 0 for lanes 0-15, 2 for 16-31

    v8f acc[NT];
#pragma unroll
    for (int n = 0; n < NT; ++n)
        acc[n] = (v8f){0.f, 0.f, 0.f, 0.f, 0.f, 0.f, 0.f, 0.f};

    const float* __restrict__ arow = A + (size_t)(mbase + nrow) * K;

    for (int k0 = 0; k0 < K; k0 += 4) {
        // A 16x4 fragment (ISA 7.12.2 32-bit A layout)
        v2f a;
        a.x = arow[k0 + koff];
        a.y = arow[k0 + koff + 1];
#pragma unroll
        for (int n = 0; n < NT; ++n) {
            const int col  = n * 16 + nrow;
            // Clamp (not mask) OOB columns: garbage only feeds C columns >= N,
            // which are never stored. Keeps EXEC all-1s around the WMMA.
            const int ccol = (col < N) ? col : (N - 1);
            v2f b;
            b.x = B[(size_t)(k0 + koff) * N + ccol];
            b.y = B[(size_t)(k0 + koff + 1) * N + ccol];
            acc[n] = __builtin_amdgcn_wmma_f32_16x16x4_f32(
                false, a, false, b, (short)0, acc[n], false, false);
        }
    }

    // C/D layout: VGPR r -> M = r (lanes 0-15) or 8+r (lanes 16-31), N = lane%16
    const int rbase = mbase + ((lane >> 4) << 3);
#pragma unroll
    for (int n = 0; n < NT; ++n) {
        const int col = n * 16 + nrow;
        if (col < N) {
#pragma unroll
            for (int r = 0; r < 8; ++r)
                C[(size_t)(rbase + r) * N + col] = acc[n][r];
        }
    }
}

// ---------------- self-loop term + bias: out = h * dis^2 + b ----------------

__global__ void vgae_self_bias(const float* __restrict__ h,
                               const float* __restrict__ dis,
                               const float* __restrict__ b,
                               float* __restrict__ out, int n, int c) {
    int idx = blockIdx.x * blockDim.x + threadIdx.x;
    if (idx < n * c) {
        int i  = idx / c;
        int ch = idx - i * c;
        float d = dis[i];
        out[idx] = h[idx] * d * d + b[ch];
    }
}

// ---------------- edge aggregation: out[dst] += h[src] * norm ---------------
// One wave per edge: src/dst/norm broadcast-loaded, lanes sweep channels.

__global__ void vgae_edge_scatter(const float* __restrict__ h,
                                  const float* __restrict__ dis,
                                  const int* __restrict__ src,
                                  const int* __restrict__ dst,
                                  float* __restrict__ out, int e, int c) {
    const int wave = threadIdx.x >> 5;
    const int lane = threadIdx.x & 31;
    const int eid  = blockIdx.x * (blockDim.x >> 5) + wave;
    if (eid >= e) return;
    const int s = src[eid];
    const int d = dst[eid];
    const float norm = dis[s] * dis[d];
    const float* __restrict__ hs = h + (size_t)s * c;
    float* od = out + (size_t)d * c;
    for (int ch = lane; ch < c; ch += 32)
        atomicAdd(&od[ch], hs[ch] * norm);
}

__global__ void vgae_relu(float* x, int n) {
    int i = blockIdx.x * blockDim.x + threadIdx.x;
    if (i < n) x[i] = fmaxf(x[i], 0.0f);
}

// ---------------------------------------------------------------------------

extern "C" void kernel_launch(void* const* d_in, const int* in_sizes, int n_in,
                              void* d_out, int out_size, void* d_ws, size_t ws_size,
                              hipStream_t stream) {
    const float* x   = (const float*)d_in[0];
    const int*   ei  = (const int*)d_in[1];
    const float* W1  = (const float*)d_in[2];
    const float* b1  = (const float*)d_in[3];
    const float* Wmu = (const float*)d_in[4];
    const float* bmu = (const float*)d_in[5];
    const float* Wls = (const float*)d_in[6];
    const float* bls = (const float*)d_in[7];

    const int N = in_sizes[0] / CH_IN;  // 100000 nodes
    const int E = in_sizes[1] / 2;      // 800000 edges
    const int* src = ei;
    const int* dst = ei + E;

    float* mu = (float*)d_out;
    float* ls = mu + (size_t)N * CH_OUT;

    // Workspace layout (~80.5 MB):
    //   dis : N floats (also holds degree during build)
    //   t1  : N*CH_HID floats (raw GEMM output; reused for tmu/tls later)
    //   h1  : N*CH_HID floats (layer-1 aggregation result)
    float* dis = (float*)d_ws;
    size_t off = ((size_t)N + 255) & ~(size_t)255;
    float* t1  = dis + off;
    float* h1  = t1 + (size_t)N * CH_HID;
    float* tmu = t1;                           // t1 dead after layer-1 scatter
    float* tls = t1 + (size_t)N * CH_OUT;

    const int thr = 256;
    const int mtiles  = (N + 15) / 16;
    const int gblocks = (mtiles + 3) / 4;      // 4 waves / block
    const int eblocks = (E + 7) / 8;           // 8 waves / block, 1 edge / wave

    // GCN normalization: deg = 1 + sum over incoming edges; dis = rsqrt(deg)
    vgae_init_deg<<<(N + thr - 1) / thr, thr, 0, stream>>>(dis, N);
    vgae_accum_deg<<<(E + thr - 1) / thr, thr, 0, stream>>>(dst, dis, E);
    vgae_rsqrt<<<(N + thr - 1) / thr, thr, 0, stream>>>(dis, N);

    // Layer 1: h1 = relu( scatter(t1[src]*norm -> dst) + t1*dis^2 + b1 ),  t1 = x @ W1
    vgae_gemm_wmma_f32<7><<<gblocks, 128, 0, stream>>>(x, W1, t1, N, CH_HID, CH_IN);
    vgae_self_bias<<<(N * CH_HID + thr - 1) / thr, thr, 0, stream>>>(t1, dis, b1, h1, N, CH_HID);
    vgae_edge_scatter<<<eblocks, thr, 0, stream>>>(t1, dis, src, dst, h1, E, CH_HID);
    vgae_relu<<<(N * CH_HID + thr - 1) / thr, thr, 0, stream>>>(h1, N * CH_HID);

    // mu / logstd heads
    vgae_gemm_wmma_f32<4><<<gblocks, 128, 0, stream>>>(h1, Wmu, tmu, N, CH_OUT, CH_HID);
    vgae_gemm_wmma_f32<4><<<gblocks, 128, 0, stream>>>(h1, Wls, tls, N, CH_OUT, CH_HID);
    vgae_self_bias<<<(N * CH_OUT + thr - 1) / thr, thr, 0, stream>>>(tmu, dis, bmu, mu, N, CH_OUT);
    vgae_self_bias<<<(N * CH_OUT + thr - 1) / thr, thr, 0, stream>>>(tls, dis, bls, ls, N, CH_OUT);
    vgae_edge_scatter<<<eblocks, thr, 0, stream>>>(tmu, dis, src, dst, mu, E, CH_OUT);
    vgae_edge_scatter<<<eblocks, thr, 0, stream>>>(tls, dis, src, dst, ls, E, CH_OUT);
}